// TranslateCube_7550552507229
// MI455X (gfx1250) — compile-verified
//
#include <hip/hip_runtime.h>
#include <cstdint>

#define IMG_H 256
#define IMG_W 256

// 8 wave32s per block; each wave produces one full 256-px output row of one
// image and stages its two 1KB source rows into a wave-private LDS slice via
// gfx1250 async global->LDS copies (ASYNCcnt path). No cross-wave sharing,
// so no workgroup barrier is needed -- just s_wait_asynccnt 0.
//
// Memory-bound roofline: 256MB in + 256MB out @ 23.3 TB/s ~= 22us; arithmetic
// (~8 flop/px) is negligible, so no WMMA formulation makes sense here. Output
// stores are non-temporal (write-once data) so the input row re-reads between
// adjacent row-blocks stay resident in the 192MB L2.
__global__ __launch_bounds__(256) void TranslateCube_kernel(
    const float* __restrict__ img,
    const float* __restrict__ dx,
    const float* __restrict__ dy,
    float* __restrict__ out,
    int n_images)
{
    __shared__ float smem[8 * 2 * IMG_W];   // 16 KB: [wave][2 rows][W]

    const int lane = threadIdx.x & 31;
    const int wave = threadIdx.x >> 5;

    const int n = blockIdx.x >> 5;                 // image id (32 blocks of 8 rows each)
    const int y = ((blockIdx.x & 31) << 3) + wave; // output row
    if (n >= n_images) return;

    const float tx = dx[n];
    const float ty = dy[n];

    // tfa.image.translate convention: out(y,x) = in(y - ty, x - tx)
    const float sy  = (float)y - ty;
    const float y0f = floorf(sy);
    const float wy  = sy - y0f;
    const int   r0  = (int)y0f;
    const int   r1  = r0 + 1;

    float* row0 = &smem[(wave * 2) * IMG_W];
    float* row1 = row0 + IMG_W;

    const size_t imgBase = (size_t)n * (size_t)(IMG_H * IMG_W);

    // ---- stage source row r0 (async DMA to LDS, or zero-fill if OOB) ----
    if (r0 >= 0 && r0 < IMG_H) {
        uint32_t l = (uint32_t)(size_t)row0 + (uint32_t)(lane << 4);
        uint64_t g = (uint64_t)(size_t)(img + imgBase + (size_t)r0 * IMG_W)
                   + (uint64_t)(lane << 4);
        // Each lane moves 16B; the instruction offset is added to BOTH the
        // global and LDS addresses, so offset:512 covers the 2nd half-row.
        asm volatile(
            "global_load_async_to_lds_b128 %0, %1, off\n\t"
            "global_load_async_to_lds_b128 %0, %1, off offset:512"
            :: "v"(l), "v"(g) : "memory");
    } else {
#pragma unroll
        for (int j = 0; j < 8; ++j) row0[lane + 32 * j] = 0.0f;
    }

    // ---- stage source row r1 ----
    if (r1 >= 0 && r1 < IMG_H) {
        uint32_t l = (uint32_t)(size_t)row1 + (uint32_t)(lane << 4);
        uint64_t g = (uint64_t)(size_t)(img + imgBase + (size_t)r1 * IMG_W)
                   + (uint64_t)(lane << 4);
        asm volatile(
            "global_load_async_to_lds_b128 %0, %1, off\n\t"
            "global_load_async_to_lds_b128 %0, %1, off offset:512"
            :: "v"(l), "v"(g) : "memory");
    } else {
#pragma unroll
        for (int j = 0; j < 8; ++j) row1[lane + 32 * j] = 0.0f;
    }

    // Wave-private staging: only ASYNCcnt ordering required, no barrier.
    asm volatile("s_wait_asynccnt 0x0" ::: "memory");

    const float omwy = 1.0f - wy;
    float* o = out + imgBase + (size_t)y * IMG_W;

#pragma unroll
    for (int i = 0; i < 8; ++i) {
        const int   x   = lane + 32 * i;           // stride-32: conflict-free LDS,
        const float sx  = (float)x - tx;           // coalesced 128B stores per wave
        const float x0f = floorf(sx);
        const float wx  = sx - x0f;
        const int   c0  = (int)x0f;
        const int   c1  = c0 + 1;

        const bool  v0 = (c0 >= 0) && (c0 < IMG_W);
        const bool  v1 = (c1 >= 0) && (c1 < IMG_W);
        const int   k0 = v0 ? c0 : 0;              // clamp + mask == clip + where
        const int   k1 = v1 ? c1 : 0;
        const float m0 = v0 ? 1.0f : 0.0f;
        const float m1 = v1 ? 1.0f : 0.0f;

        const float a0 = row0[k0] * m0;            // c00 (row validity already
        const float a1 = row0[k1] * m1;            //  folded in via zero-fill)
        const float b0 = row1[k0] * m0;            // c10
        const float b1 = row1[k1] * m1;            // c11

        const float top = (1.0f - wx) * a0 + wx * a1;
        const float bot = (1.0f - wx) * b0 + wx * b1;
        // Write-once output: non-temporal store (th:TH_STORE_NT) keeps the
        // 192MB L2 free for input-row reuse between adjacent row-blocks.
        __builtin_nontemporal_store(omwy * top + wy * bot, &o[x]);
    }
}

extern "C" void kernel_launch(void* const* d_in, const int* in_sizes, int n_in,
                              void* d_out, int out_size, void* d_ws, size_t ws_size,
                              hipStream_t stream) {
    const float* images = (const float*)d_in[0]; // [B*T, 256, 256] f32
    const float* dx     = (const float*)d_in[1]; // [B*T] f32
    const float* dy     = (const float*)d_in[2]; // [B*T] f32
    // d_in[3] = winsize (unused by translate, as in the reference)
    float* out = (float*)d_out;

    const int n_images = in_sizes[1];            // B*T (= 1024)
    dim3 block(256);                             // 8 wave32s
    dim3 grid((unsigned)n_images * (IMG_H / 8)); // one block per 8 output rows
    TranslateCube_kernel<<<grid, block, 0, stream>>>(images, dx, dy, out, n_images);
}